// GCN_67156108640501
// MI455X (gfx1250) — compile-verified
//
#include <hip/hip_runtime.h>
#include <math.h>

typedef __attribute__((ext_vector_type(2))) float v2f;
typedef __attribute__((ext_vector_type(8))) float v8f;

#define DIN 128
#define PAD 16

// ---------------- degree / norm ----------------
__global__ void k_init_deg(float* __restrict__ deg, int n) {
    int i = blockIdx.x * blockDim.x + threadIdx.x;
    if (i < n) deg[i] = 1.0f;               // self loop contributes 1
}

__global__ void k_deg_accum(const int* __restrict__ dst, float* __restrict__ deg, int e) {
    int i = blockIdx.x * blockDim.x + threadIdx.x;
    if (i < e) atomicAdd(&deg[dst[i]], 1.0f);
}

__global__ void k_rsqrt_inplace(float* __restrict__ deg, int n) {
    int i = blockIdx.x * blockDim.x + threadIdx.x;
    if (i < n) deg[i] = rsqrtf(deg[i]);
}

// ---------------- layer-1 GEMM: T[N,16] = X[N,128] @ pad(W1)[128,16] ----------------
// One wave32 per 16-row tile; V_WMMA_F32_16X16X4_F32, K=128 fully unrolled (32 WMMAs).
// Out-of-range rows load a clamped row unconditionally (their D rows are never
// stored), keeping the inner loop branch-free so loads pipeline ahead of WMMAs.
__global__ __launch_bounds__(256) void k_gemm_wmma(const float* __restrict__ X,
                                                   const float* __restrict__ W,
                                                   float* __restrict__ T, int n) {
    __shared__ float sW[DIN * PAD];         // 8 KB: W1 zero-padded 128x16
    for (int t = threadIdx.x; t < DIN * PAD; t += 256) {
        int kk = t >> 4, nn = t & 15;
        sW[t] = (nn < 10) ? W[kk * 10 + nn] : 0.0f;
    }
    __syncthreads();

    const int wave = threadIdx.x >> 5;
    const int lane = threadIdx.x & 31;
    const int base = (blockIdx.x * 8 + wave) * 16;
    const int half = lane >> 4;             // 0: K=0,1   1: K=2,3
    const int m    = lane & 15;             // A row / B col / D col
    const int kb   = half * 2;

    int rowA = base + m;
    if (rowA > n - 1) rowA = n - 1;         // clamp: harmless, rows not stored
    const float* xr = X + (size_t)rowA * DIN + kb;
    const float* bw = sW + kb * PAD + m;

    v8f c = {};
    #pragma unroll
    for (int k = 0; k < DIN; k += 4) {
        v2f a;
        a.x = xr[k];
        a.y = xr[k + 1];
        v2f b;
        b.x = bw[k * PAD];
        b.y = bw[k * PAD + PAD];
        c = __builtin_amdgcn_wmma_f32_16x16x4_f32(false, a, false, b,
                                                  (short)0, c, false, false);
    }

    float* to = T + (size_t)(base + half * 8) * PAD + m;
    if (base + 16 <= n) {                   // full tile: uniform, unguarded stores
        #pragma unroll
        for (int j = 0; j < 8; ++j) to[(size_t)j * PAD] = c[j];
    } else {
        #pragma unroll
        for (int j = 0; j < 8; ++j) {
            int row = base + j + half * 8;
            if (row < n) T[(size_t)row * PAD + m] = c[j];
        }
    }
}

// ---------------- tiny per-node transform: T[:,0:FOUT] = H[:,0:FIN] @ W ----------------
template <int FIN, int FOUT>
__global__ void k_transform(const float* __restrict__ H, const float* __restrict__ W,
                            float* __restrict__ T, int n) {
    int i = blockIdx.x * blockDim.x + threadIdx.x;
    if (i >= n) return;
    float in[FIN];
    #pragma unroll
    for (int k = 0; k < FIN; ++k) in[k] = H[(size_t)i * PAD + k];
    #pragma unroll
    for (int f = 0; f < FOUT; ++f) {
        float acc = 0.0f;
        #pragma unroll
        for (int k = 0; k < FIN; ++k) acc += in[k] * W[k * FOUT + f];
        T[(size_t)i * PAD + f] = acc;
    }
}

// ---------------- agg init with fused self-loop: agg = T * dinv[i]^2 ----------------
template <int F>
__global__ void k_agg_init(const float* __restrict__ T, const float* __restrict__ dinv,
                           float* __restrict__ agg, int n) {
    int i = blockIdx.x * blockDim.x + threadIdx.x;
    if (i >= n) return;
    float w = dinv[i] * dinv[i];
    #pragma unroll
    for (int f = 0; f < F; ++f) agg[(size_t)i * PAD + f] = T[(size_t)i * PAD + f] * w;
}

// ---------------- edge scatter: agg[dst] += T[src] * dinv[src]*dinv[dst] ----------------
template <int F>
__global__ void k_scatter(const int* __restrict__ src, const int* __restrict__ dst,
                          const float* __restrict__ dinv, const float* __restrict__ T,
                          float* __restrict__ agg, int e) {
    int i = blockIdx.x * blockDim.x + threadIdx.x;
    if (i >= e) return;
    int s = src[i], d = dst[i];
    float w = dinv[s] * dinv[d];
    const float* ts = T + (size_t)s * PAD;
    float* ad = agg + (size_t)d * PAD;
    #pragma unroll
    for (int f = 0; f < F; ++f) atomicAdd(&ad[f], ts[f] * w);
}

// ---------------- bias + tanh ----------------
template <int F>
__global__ void k_finalize(const float* __restrict__ agg, const float* __restrict__ b,
                           float* __restrict__ H, int n) {
    int i = blockIdx.x * blockDim.x + threadIdx.x;
    if (i >= n) return;
    #pragma unroll
    for (int f = 0; f < F; ++f)
        H[(size_t)i * PAD + f] = tanhf(agg[(size_t)i * PAD + f] + b[f]);
}

// ---------------- head: out = h @ Wc + bc ; emit (out, h) concatenated ----------------
__global__ void k_head(const float* __restrict__ H, const float* __restrict__ Wc,
                       const float* __restrict__ bc, float* __restrict__ out, int n) {
    int i = blockIdx.x * blockDim.x + threadIdx.x;
    if (i >= n) return;
    float h0 = H[(size_t)i * PAD + 0];
    float h1 = H[(size_t)i * PAD + 1];
    float o0 = h0 * Wc[0] + h1 * Wc[2] + bc[0];
    float o1 = h0 * Wc[1] + h1 * Wc[3] + bc[1];
    out[(size_t)i * 2 + 0] = o0;
    out[(size_t)i * 2 + 1] = o1;
    out[(size_t)2 * n + (size_t)i * 2 + 0] = h0;
    out[(size_t)2 * n + (size_t)i * 2 + 1] = h1;
}

extern "C" void kernel_launch(void* const* d_in, const int* in_sizes, int n_in,
                              void* d_out, int out_size, void* d_ws, size_t ws_size,
                              hipStream_t stream) {
    const float* x  = (const float*)d_in[0];
    const int*   ei = (const int*)d_in[1];          // edge_index [2,E] int32
    const float* W1 = (const float*)d_in[2];
    const float* b1 = (const float*)d_in[3];
    const float* W2 = (const float*)d_in[4];
    const float* b2 = (const float*)d_in[5];
    const float* W3 = (const float*)d_in[6];
    const float* b3 = (const float*)d_in[7];
    const float* Wc = (const float*)d_in[8];
    const float* bc = (const float*)d_in[9];

    const int N = in_sizes[0] / DIN;
    const int E = in_sizes[1] / 2;
    const int* src = ei;
    const int* dst = ei + E;

    float* ws   = (float*)d_ws;
    float* dinv = ws;                       // N   (deg, then rsqrt in place)
    float* T    = dinv + N;                 // N*16 transform output
    float* AGG  = T + (size_t)N * PAD;      // N*16 accumulator
    float* H    = AGG + (size_t)N * PAD;    // N*16 activations

    const int TPB = 256;
    const int gN  = (N + TPB - 1) / TPB;
    const int gE  = (E + TPB - 1) / TPB;

    // degree + symmetric norm
    k_init_deg<<<gN, TPB, 0, stream>>>(dinv, N);
    k_deg_accum<<<gE, TPB, 0, stream>>>(dst, dinv, E);
    k_rsqrt_inplace<<<gN, TPB, 0, stream>>>(dinv, N);

    // layer 1: WMMA transform, then aggregate (F=10)
    const int tiles = (N + 15) / 16;
    k_gemm_wmma<<<(tiles + 7) / 8, TPB, 0, stream>>>(x, W1, T, N);
    k_agg_init<10><<<gN, TPB, 0, stream>>>(T, dinv, AGG, N);
    k_scatter<10><<<gE, TPB, 0, stream>>>(src, dst, dinv, T, AGG, E);
    k_finalize<10><<<gN, TPB, 0, stream>>>(AGG, b1, H, N);

    // layer 2 (10 -> 5)
    k_transform<10, 5><<<gN, TPB, 0, stream>>>(H, W2, T, N);
    k_agg_init<5><<<gN, TPB, 0, stream>>>(T, dinv, AGG, N);
    k_scatter<5><<<gE, TPB, 0, stream>>>(src, dst, dinv, T, AGG, E);
    k_finalize<5><<<gN, TPB, 0, stream>>>(AGG, b2, H, N);

    // layer 3 (5 -> 2)
    k_transform<5, 2><<<gN, TPB, 0, stream>>>(H, W3, T, N);
    k_agg_init<2><<<gN, TPB, 0, stream>>>(T, dinv, AGG, N);
    k_scatter<2><<<gE, TPB, 0, stream>>>(src, dst, dinv, T, AGG, E);
    k_finalize<2><<<gN, TPB, 0, stream>>>(AGG, b3, H, N);

    // classifier head + emit (out, h)
    k_head<<<gN, TPB, 0, stream>>>(H, Wc, bc, (float*)d_out, N);
}